// PLILayer_65962107732068
// MI455X (gfx1250) — compile-verified
//
#include <hip/hip_runtime.h>
#include <hip/hip_bf16.h>
#include <math.h>

// Problem dims (from reference setup_inputs).
#define BB 16
#define CC 64
#define SS 1024          // FFT length N == samples S
#define RR (BB * CC)     // 1024 rows total

typedef __attribute__((ext_vector_type(2))) float v2f;
typedef __attribute__((ext_vector_type(8))) float v8f;

// ---------------------------------------------------------------------------
// Kernel 1: discrete Hilbert circulant kernel, closed form for even N:
//   k[m] = (2/N) * cot(pi*m/N)  for odd m,  0 for even m (incl. m=0, N/2).
// Then Im(analytic)[n] = sum_m k[(n-m) mod N] * x[m]; Re(analytic) = x.
// ---------------------------------------------------------------------------
__global__ void hilbert_kvec_kernel(float* __restrict__ kvec) {
    int m = blockIdx.x * blockDim.x + threadIdx.x;
    if (m < SS) {
        float v = 0.0f;
        if (m & 1) {
            float ang = 3.14159265358979323846f * (float)m * (1.0f / (float)SS);
            v = (2.0f / (float)SS) / tanf(ang);
        }
        kvec[m] = v;
    }
}

// ---------------------------------------------------------------------------
// Kernel 2: Y = X @ Kt, Kt[m][n] = kvec[(n - m) mod N]  (circulant GEMM).
//
// Block = 4 waves (128 threads). Each wave owns a 64x16 output strip:
// 4 row-tiles stacked in M sharing ONE circulant B fragment per K-step
// (B depends only on n - k), -> 4x v_wmma_f32_16x16x4_f32 per B gather.
//
// LDS holds the REVERSED kernel klds[m] = kvec[(-m) mod N] (+1 wrap pad), so
// the per-lane B element index is (kk + khalf - ncol) mod N, which ascends by
// 4 per K-step: one add + one mask, two ds loads at idx / idx+1.
//
// ISA 7.12.2 fragment layouts (wave32):
//   A 16x4 f32: lanes 0-15 -> M=lane, v0=K0, v1=K1; lanes 16-31 -> M=lane-16,
//               v0=K2, v1=K3.
//   B 4x16 f32: N on lanes, same K split per half-wave.
//   C/D 16x16 f32: VGPR i: lanes 0-15 -> M=i, N=lane; lanes 16-31 -> M=8+i.
// ---------------------------------------------------------------------------
__global__ __launch_bounds__(128) void hilbert_gemm_wmma(
    const float* __restrict__ X,      // [RR, SS]
    const float* __restrict__ kvec,   // [SS]
    float* __restrict__ Y) {          // [RR, SS]
    __shared__ float klds[SS + 4];    // reversed kernel + wrap pad

    const int tid  = threadIdx.x;
    const int lane = tid & 31;
    const int wv   = tid >> 5;        // wave id 0..3

    // Cooperative staging of the reversed circulant kernel.
    for (int m = tid; m < SS; m += 128) klds[m] = kvec[(SS - m) & (SS - 1)];
    if (tid == 0) klds[SS] = klds[0];             // wrap pad for idx+1
    __syncthreads();

    const int r0 = blockIdx.x * 64;               // 64-row strip (4 M-tiles)
    const int n0 = (blockIdx.y * 4 + wv) * 16;    // this wave's column tile

    const int sub   = lane & 15;
    const int khalf = (lane >> 4) << 1;           // 0 or 2 (K half per ISA)
    const float* arow = X + (size_t)(r0 + sub) * SS + khalf;
    const int ncol = n0 + sub;

    // B element index for K = khalf at kk = 0 (ascends by 4 per K-step).
    int bidx = (khalf - ncol) & (SS - 1);

    v8f acc0 = {}, acc1 = {}, acc2 = {}, acc3 = {};
    for (int kk = 0; kk < SS; kk += 4) {
        // One circulant B fragment, shared by all 4 row tiles.
        v2f b;
        b.x = klds[bidx];
        b.y = klds[bidx + 1];                     // pad covers bidx == SS-1
        bidx = (bidx + 4) & (SS - 1);

        // Four A fragments (consecutive f32 pairs -> b64 loads).
        v2f a0, a1, a2, a3;
        a0.x = arow[kk];               a0.y = arow[kk + 1];
        a1.x = arow[16 * SS + kk];     a1.y = arow[16 * SS + kk + 1];
        a2.x = arow[32 * SS + kk];     a2.y = arow[32 * SS + kk + 1];
        a3.x = arow[48 * SS + kk];     a3.y = arow[48 * SS + kk + 1];

        acc0 = __builtin_amdgcn_wmma_f32_16x16x4_f32(false, a0, false, b,
                                                     (short)0, acc0, false, false);
        acc1 = __builtin_amdgcn_wmma_f32_16x16x4_f32(false, a1, false, b,
                                                     (short)0, acc1, false, false);
        acc2 = __builtin_amdgcn_wmma_f32_16x16x4_f32(false, a2, false, b,
                                                     (short)0, acc2, false, false);
        acc3 = __builtin_amdgcn_wmma_f32_16x16x4_f32(false, a3, false, b,
                                                     (short)0, acc3, false, false);
    }

    // Store the four 16x16 tiles (C/D layout per ISA 7.12.2).
    const int rb = r0 + ((lane >> 4) ? 8 : 0);
#pragma unroll
    for (int i = 0; i < 8; ++i) {
        Y[(size_t)(rb + i) * SS + ncol]      = acc0[i];
        Y[(size_t)(rb + 16 + i) * SS + ncol] = acc1[i];
        Y[(size_t)(rb + 32 + i) * SS + ncol] = acc2[i];
        Y[(size_t)(rb + 48 + i) * SS + ncol] = acc3[i];
    }
}

// ---------------------------------------------------------------------------
// Kernel 3: PLI. One wave per (b, i, j):
//   pli = |mean_s sign(Yi*Xj - Xi*Yj)|, diag -> 0, threshold.
// sign(cross) == sign(sin(theta_i - theta_j)) == reference's sign(wrapped d),
// including the |w|==pi -> 0 case (cross product is exactly 0 there).
// ---------------------------------------------------------------------------
__global__ __launch_bounds__(256) void pli_pairs_kernel(
    const float* __restrict__ X,      // Re(analytic) = x, [BB*CC, SS]
    const float* __restrict__ Y,      // Im(analytic),     [BB*CC, SS]
    const int* __restrict__ thr,
    float* __restrict__ out) {        // [BB, CC, CC]
    const int wave = (int)((blockIdx.x * blockDim.x + threadIdx.x) >> 5);
    const int lane = threadIdx.x & 31;
    if (wave >= BB * CC * CC) return;

    const int j = wave & (CC - 1);
    const int i = (wave >> 6) & (CC - 1);
    const int b = wave >> 12;

    const float* xi = X + (size_t)(b * CC + i) * SS;
    const float* yi = Y + (size_t)(b * CC + i) * SS;
    const float* xj = X + (size_t)(b * CC + j) * SS;
    const float* yj = Y + (size_t)(b * CC + j) * SS;

    float s = 0.0f;
    for (int t = lane; t < SS; t += 32) {
        float cp = yi[t] * xj[t] - xi[t] * yj[t];
        s += (cp > 0.0f) ? 1.0f : ((cp < 0.0f) ? -1.0f : 0.0f);
    }
    // wave32 reduction
    for (int off = 16; off > 0; off >>= 1) s += __shfl_down(s, off, 32);

    if (lane == 0) {
        float pli = fabsf(s * (1.0f / (float)SS));
        if (i == j) pli = 0.0f;
        float th = (float)thr[0];
        out[wave] = (pli >= th) ? pli : 0.0f;
    }
}

// ---------------------------------------------------------------------------
// Host launcher (graph-capture safe: only kernel launches on `stream`).
// Workspace layout: [0, 4KB) kvec | [4KB, 4KB + 4MB) Y.
// ---------------------------------------------------------------------------
extern "C" void kernel_launch(void* const* d_in, const int* in_sizes, int n_in,
                              void* d_out, int out_size, void* d_ws, size_t ws_size,
                              hipStream_t stream) {
    (void)in_sizes; (void)n_in; (void)out_size; (void)ws_size;

    const float* x  = (const float*)d_in[0];   // [16, 64, 1024] f32
    const int* thr  = (const int*)d_in[1];     // scalar int
    float* out      = (float*)d_out;           // [16, 64, 64] f32

    float* kvec = (float*)d_ws;                // 1024 f32
    float* Yim  = kvec + SS;                   // 1024*1024 f32

    hilbert_kvec_kernel<<<(SS + 255) / 256, 256, 0, stream>>>(kvec);

    dim3 gemm_grid(RR / 64, SS / (16 * 4));    // 16 x 16 blocks, 4 waves each
    hilbert_gemm_wmma<<<gemm_grid, 128, 0, stream>>>(x, kvec, Yim);

    const int total_waves = BB * CC * CC;      // 65536 pairs
    pli_pairs_kernel<<<(total_waves * 32) / 256, 256, 0, stream>>>(x, Yim, thr, out);
}